// GraphConvolution_22041772163509
// MI455X (gfx1250) — compile-verified
//
#include <hip/hip_runtime.h>

// GCN layer for MI455X (gfx1250, wave32):
//   support = x @ W      fp32 WMMA 16x16x4; one wave per 16x128 row strip,
//                        W staged transposed in LDS (stride-132 padding ->
//                        conflict-free ds_load_b64 B operands).
//   out[r] += v*support[c]  one wave per edge, float4/lane gather, fp32 L2 atomics.

typedef float v2f __attribute__((ext_vector_type(2)));
typedef float v8f __attribute__((ext_vector_type(8)));

#define FEAT 128            // D
#define LDS_STRIDE 132      // 128 + 4 floats padding: bank = (4n + k) % 64 -> conflict-free

// ---------------------------------------------------------------------------
// Dense GEMM: support[N,128] = x[N,128] @ W[128,128].
// Block = 256 threads = 8 waves; each wave owns a 16-row strip and computes
// all 8 N-tiles (8 x v8f accumulators). W is loaded once per block into LDS,
// transposed: ldsW[n*132 + k] = W[k][n].
//
// WMMA operand layouts (CDNA5 ISA 7.12.2), lane = m/n in [0,16), khalf = lane>>4:
//   A (16x4):  {v0,v1} = A[m][k0], A[m][k0+1],   k0 = ks + 2*khalf
//   B (4x16):  {v0,v1} = B[k0][n], B[k0+1][n]    (contiguous in transposed LDS)
//   C/D:       8 VGPRs; N = lane&15, M = vgpr + 8*khalf
// ---------------------------------------------------------------------------
__global__ __launch_bounds__(256) void gcn_gemm_wmma_f32(
    const float* __restrict__ x, const float* __restrict__ w,
    float* __restrict__ support, int mTiles)
{
    __shared__ float ldsW[FEAT * LDS_STRIDE];   // 66 KB of the 320 KB WGP LDS

    // Stage W transposed into LDS (coalesced global reads, one-time cost).
    for (int i = threadIdx.x; i < FEAT * FEAT; i += 256) {
        const int k = i >> 7;       // row of W
        const int n = i & 127;      // col of W
        ldsW[n * LDS_STRIDE + k] = w[i];
    }
    __syncthreads();

    const int strip = blockIdx.x * 8 + (threadIdx.x >> 5);   // 16-row strip id
    if (strip >= mTiles) return;

    const int lane  = threadIdx.x & 31;
    const int mn    = lane & 15;     // M index for A, N index for B/C/D
    const int khalf = lane >> 4;     // 0 or 1

    const float* arow = x + (size_t)(strip * 16 + mn) * FEAT;

    v8f acc[8];
#pragma unroll
    for (int nt = 0; nt < 8; ++nt) acc[nt] = (v8f){};

    for (int ks = 0; ks < FEAT; ks += 4) {
        const int k0 = ks + khalf * 2;
        const v2f a = *(const v2f*)(arow + k0);              // A[m][k0..k0+1]
#pragma unroll
        for (int nt = 0; nt < 8; ++nt) {
            // B[k0..k0+1][nt*16 + n] -> one ds_load_b64, bank-conflict-free
            const v2f b = *(const v2f*)(&ldsW[(nt * 16 + mn) * LDS_STRIDE + k0]);
            acc[nt] = __builtin_amdgcn_wmma_f32_16x16x4_f32(
                false, a, false, b, (short)0, acc[nt], false, false);
        }
    }

    float* orow = support + (size_t)(strip * 16) * FEAT;
    const int mbase = khalf * 8;
#pragma unroll
    for (int nt = 0; nt < 8; ++nt)
#pragma unroll
        for (int j = 0; j < 8; ++j)
            orow[(size_t)(mbase + j) * FEAT + nt * 16 + mn] = acc[nt][j];
}

// ---------------------------------------------------------------------------
// COO SpMM: out[row[e]] += val[e] * support[col[e]], one wave32 per edge.
// Lane l owns features [4l, 4l+4): one global_load_b128 gather (512 B
// contiguous per wave; support is L2-resident: 51.2 MB vs 192 MB L2) and
// 4 hardware fp32 atomic adds (no-return, stays in L2). 128 atomics/edge is
// the minimum for unsorted COO.
// ---------------------------------------------------------------------------
__global__ __launch_bounds__(256) void gcn_spmm_atomic(
    const float* __restrict__ support,
    const float* __restrict__ edge_vals,
    const int*   __restrict__ edge_row,
    const int*   __restrict__ edge_col,
    float* __restrict__ out, int E)
{
    const int wave = blockIdx.x * (blockDim.x >> 5) + (threadIdx.x >> 5);
    const int lane = threadIdx.x & 31;
    if (wave >= E) return;

    const int   r = __builtin_amdgcn_readfirstlane(edge_row[wave]);
    const int   c = __builtin_amdgcn_readfirstlane(edge_col[wave]);
    const float v = edge_vals[wave];

    const float4 s = ((const float4*)(support + (size_t)c * FEAT))[lane];

    float* dst = out + (size_t)r * FEAT + lane * 4;
    unsafeAtomicAdd(dst + 0, v * s.x);
    unsafeAtomicAdd(dst + 1, v * s.y);
    unsafeAtomicAdd(dst + 2, v * s.z);
    unsafeAtomicAdd(dst + 3, v * s.w);
}

extern "C" void kernel_launch(void* const* d_in, const int* in_sizes, int n_in,
                              void* d_out, int out_size, void* d_ws, size_t ws_size,
                              hipStream_t stream)
{
    const float* x         = (const float*)d_in[0];   // [N,128]
    const float* w         = (const float*)d_in[1];   // [128,128]
    const float* edge_vals = (const float*)d_in[2];   // [E]
    const int*   edge_row  = (const int*)d_in[3];     // [E]
    const int*   edge_col  = (const int*)d_in[4];     // [E]
    float*       out       = (float*)d_out;           // [N,128]
    float*       support   = (float*)d_ws;            // [N,128] scratch (51.2 MB)

    const int N = in_sizes[0] / FEAT;                 // 100000 (divisible by 16)
    const int E = in_sizes[2];                        // 1600000

    // segment_sum accumulates into out -> zero it (graph-capturable memset).
    hipMemsetAsync(d_out, 0, (size_t)out_size * sizeof(float), stream);

    // GEMM: one wave per 16x128 strip; 8 strips (128 rows) per block.
    const int mTiles = (N + 15) / 16;                 // 6250 strips
    gcn_gemm_wmma_f32<<<dim3((mTiles + 7) / 8), dim3(256), 0, stream>>>(
        x, w, support, mTiles);

    // SpMM: one wave per edge; 8 edges per 256-thread block.
    gcn_spmm_atomic<<<dim3((E + 7) / 8), dim3(256), 0, stream>>>(
        support, edge_vals, edge_row, edge_col, out, E);
}